// ProjectNet_45329084842165
// MI455X (gfx1250) — compile-verified
//
#include <hip/hip_runtime.h>
#include <math.h>

// NetVLAD forward for MI455X (gfx1250, wave32).
// Both GEMMs run on V_WMMA_F32_16X16X4_F32 (fp32 matrix pipe) — inputs are
// fp32 and the problem is L2-resident, so no precision down-convert pays off.
// Phase-1 x-tile staging uses the CDNA5 async global->LDS pipe
// (GLOBAL_LOAD_ASYNC_TO_LDS_B128, ASYNCcnt-tracked, bypasses VGPRs).

typedef float v2f __attribute__((ext_vector_type(2)));
typedef float v8f __attribute__((ext_vector_type(8)));
typedef int   v4i __attribute__((ext_vector_type(4)));

#define B_  16
#define N_  1024
#define D_  512
#define K_  32
#define EPSF 1e-12f
#define XSTR 516   // LDS row stride in floats (516 % 64 = 4 -> conflict-free A reads)

#if __has_builtin(__builtin_amdgcn_global_load_async_to_lds_b128)
#define HAVE_ASYNC_LDS 1
#else
#define HAVE_ASYNC_LDS 0
#endif

#if __has_builtin(__builtin_amdgcn_s_wait_asynccnt)
#define WAIT_ASYNC0() __builtin_amdgcn_s_wait_asynccnt(0)
#else
#define WAIT_ASYNC0() asm volatile("s_wait_asynccnt 0x0" ::: "memory")
#endif

__device__ __forceinline__ float wsum(float v) {
  #pragma unroll
  for (int o = 16; o > 0; o >>= 1) v += __shfl_xor(v, o, 32);
  return v;
}
__device__ __forceinline__ float wmax(float v) {
  #pragma unroll
  for (int o = 16; o > 0; o >>= 1) v = fmaxf(v, __shfl_xor(v, o, 32));
  return v;
}

// ---------------------------------------------------------------------------
// Phase 1: per (b, 16-row n-tile): row L2 norms, logits = xn@W + b (WMMA),
// masked softmax, soft_assign out, assign_sum / entropy / mask_sum atomics.
// Block = 512 threads = 16 waves. Grid = B * N/16 = 1024.
// ---------------------------------------------------------------------------
__global__ __launch_bounds__(512)
void netvlad_phase1(const float* __restrict__ x, const float* __restrict__ mask,
                    const float* __restrict__ W, const float* __restrict__ bias,
                    float* __restrict__ sa_ws, float* __restrict__ invn_ws,
                    float* __restrict__ assign_sum, float* __restrict__ ent_acc,
                    float* __restrict__ mask_sum)
{
  __shared__ float xl[16 * XSTR];       // 16 x 512 tile, padded
  __shared__ float part[4][16][16];     // per-wave partial C tiles
  __shared__ float invl[16];

  const int tid  = threadIdx.x;
  const int wave = tid >> 5;
  const int lane = tid & 31;
  const int b    = blockIdx.x >> 6;         // N_/16 = 64 tiles per batch
  const int n0   = (blockIdx.x & 63) << 4;

  const float* xrb = x + ((size_t)b * N_ + n0) * D_;

#if HAVE_ASYNC_LDS
  // async global->LDS staging: 2048 16B quads, 4 per thread, VGPR-free.
  // Row base offsets are 16B aligned (XSTR*4 = 2064 = 129*16).
  // Builtin signature (from clang diagnostics): param0 = AS(1) v4i* (non-const
  // global source), param1 = AS(3) v4i* (LDS dest), then offset, cpol.
  #pragma unroll
  for (int j = 0; j < 4; ++j) {
    const int q   = tid + j * 512;
    const int row = q >> 7;               // q / 128
    const int c4  = (q & 127) << 2;       // 4-float column start
    __builtin_amdgcn_global_load_async_to_lds_b128(
        (__attribute__((address_space(1))) v4i*)(xrb + (size_t)row * D_ + c4),
        (__attribute__((address_space(3))) v4i*)&xl[row * XSTR + c4],
        0, 0);
  }
  WAIT_ASYNC0();
#else
  // fallback: VGPR-staged coalesced fill (thread tid = column, 16 rows)
  #pragma unroll
  for (int i = 0; i < 16; ++i)
    xl[i * XSTR + tid] = xrb[(size_t)i * D_ + tid];
#endif
  __syncthreads();

  // wave w computes inverse L2 norm of row w
  {
    const float* r = &xl[wave * XSTR];
    float s = 0.f;
    #pragma unroll
    for (int j = 0; j < 16; ++j) { float v = r[lane + 32 * j]; s += v * v; }
    s = wsum(s);
    float inv = 1.f / fmaxf(sqrtf(s), EPSF);
    if (lane == 0) {
      invl[wave] = inv;
      invn_ws[(size_t)b * N_ + n0 + wave] = inv;
    }
  }
  __syncthreads();

  // 4 waves run the logits GEMM: wave -> (ktile = w>>1, D-half = w&1)
  if (wave < 4) {
    const int kt  = wave >> 1;
    const int dlo = (wave & 1) * 256;
    const int m   = lane & 15;          // A row / B col (mod 16)
    const int hi  = lane >> 4;          // K-offset group
    const int kk  = kt * 16 + m;        // W column
    const float scale = invl[m];        // fold 1/||x|| into A
    const float* xrow = &xl[m * XSTR];
    v8f c = {};
    for (int d = dlo; d < dlo + 256; d += 4) {
      const int da = d + 2 * hi;
      v2f a, bm;
      a.x  = xrow[da]     * scale;
      a.y  = xrow[da + 1] * scale;
      bm.x = W[(size_t)da       * K_ + kk];
      bm.y = W[(size_t)(da + 1) * K_ + kk];
      c = __builtin_amdgcn_wmma_f32_16x16x4_f32(false, a, false, bm,
                                                (short)0, c, false, false);
    }
    #pragma unroll
    for (int v = 0; v < 8; ++v)
      part[wave][v + 8 * hi][m] = c[v];   // C: M = v + 8*(lane>=16), N = lane%16
  }
  __syncthreads();

  // wave w = row w, lane = k: masked softmax over K=32, entropy, sums
  {
    const int r = wave, k = lane;
    const float mrow = mask[(size_t)b * N_ + n0 + r];
    const int kt = k >> 4, kc = k & 15;
    float lg = part[2 * kt][r][kc] + part[2 * kt + 1][r][kc];
    lg = (lg + bias[k]) * mrow;
    const float mx = wmax(lg);
    const float e  = expf(lg - mx);
    const float se = wsum(e);
    const float sa = (e / se) * mrow;
    sa_ws[((size_t)b * N_ + n0 + r) * K_ + k] = sa;
    atomicAdd(&assign_sum[b * K_ + k], sa);
    float ent = (sa > 0.f) ? (-sa * log2f(sa)) : 0.f;
    ent = wsum(ent);
    if (lane == 0) {
      atomicAdd(ent_acc, ent);
      atomicAdd(mask_sum, mrow);
    }
  }
}

// ---------------------------------------------------------------------------
// Phase 2: per batch, V[d,k] = sum_n xn[n,d]*sa[n,k]  (512x1024x32 GEMM),
// then subtract centroids*assign_sum. Block = 64 thr (2 waves, one per
// K-tile); grid = B * D/16 = 512. 256 fp32 WMMAs per wave.
// ---------------------------------------------------------------------------
__global__ __launch_bounds__(64)
void netvlad_phase2(const float* __restrict__ x, const float* __restrict__ invn,
                    const float* __restrict__ sa, const float* __restrict__ cent,
                    const float* __restrict__ assign_sum,
                    float* __restrict__ vlad_ws)
{
  __shared__ float invl[N_];  // 4 KB
  const int tid  = threadIdx.x;
  const int lane = tid & 31;
  const int kt   = tid >> 5;                // K-tile 0/1
  const int b    = blockIdx.x >> 5;         // D_/16 = 32 tiles per batch
  const int d0   = (blockIdx.x & 31) << 4;

  for (int i = tid; i < N_; i += 64) invl[i] = invn[(size_t)b * N_ + i];
  __syncthreads();

  const int m  = lane & 15;
  const int hi = lane >> 4;
  const int k  = kt * 16 + m;
  const float* xb  = x  + (size_t)b * N_ * D_;
  const float* sab = sa + (size_t)b * N_ * K_;

  v8f c = {};
  for (int n = 0; n < N_; n += 4) {
    const int nn = n + 2 * hi;
    v2f a, bm;
    a.x  = xb[(size_t)nn       * D_ + d0 + m] * invl[nn];       // A = xn^T tile
    a.y  = xb[(size_t)(nn + 1) * D_ + d0 + m] * invl[nn + 1];
    bm.x = sab[(size_t)nn       * K_ + k];                      // B = sa tile
    bm.y = sab[(size_t)(nn + 1) * K_ + k];
    c = __builtin_amdgcn_wmma_f32_16x16x4_f32(false, a, false, bm,
                                              (short)0, c, false, false);
  }
  const float as = assign_sum[b * K_ + k];
  #pragma unroll
  for (int v = 0; v < 8; ++v) {
    const int d = d0 + v + 8 * hi;
    vlad_ws[((size_t)b * D_ + d) * K_ + k] = c[v] - cent[(size_t)d * K_ + k] * as;
  }
}

// ---------------------------------------------------------------------------
// Phase 3a: intra-normalize vlad columns over D, write transposed (B,K,D).
// One wave per (b,k). Grid = B*K/4 blocks of 128.
// ---------------------------------------------------------------------------
__global__ __launch_bounds__(128)
void netvlad_phase3_norm(const float* __restrict__ vlad_ws, float* __restrict__ out)
{
  const int g    = blockIdx.x * 4 + (threadIdx.x >> 5);
  const int lane = threadIdx.x & 31;
  const int b = g >> 5, k = g & 31;
  const float* col = vlad_ws + (size_t)b * D_ * K_ + k;
  float vals[16], s = 0.f;
  #pragma unroll
  for (int j = 0; j < 16; ++j) {
    float v = col[(size_t)(lane + 32 * j) * K_];
    vals[j] = v; s += v * v;
  }
  s = wsum(s);
  const float inv = 1.f / fmaxf(sqrtf(s), EPSF);
  float* orow = out + ((size_t)b * K_ + k) * D_;
  #pragma unroll
  for (int j = 0; j < 16; ++j) orow[lane + 32 * j] = vals[j] * inv;
}

// ---------------------------------------------------------------------------
// Phase 3b: assign_dist = log_softmax(assign_sum) per batch; entropy finalize.
// One block, wave per batch, lane = k.
// ---------------------------------------------------------------------------
__global__ __launch_bounds__(512)
void netvlad_phase3_dist(const float* __restrict__ assign_sum,
                         const float* __restrict__ ent_acc,
                         const float* __restrict__ mask_sum,
                         float* __restrict__ out)
{
  const int bb = threadIdx.x >> 5, k = threadIdx.x & 31;
  const float s  = assign_sum[bb * K_ + k];
  const float mx = wmax(s);
  const float se = wsum(expf(s - mx));
  out[(size_t)B_ * K_ * D_ + bb * K_ + k] = (s - mx) - logf(se);
  if (threadIdx.x == 0)
    out[(size_t)B_ * K_ * D_ + B_ * K_] = ent_acc[0] / mask_sum[0];
}

// ---------------------------------------------------------------------------
extern "C" void kernel_launch(void* const* d_in, const int* in_sizes, int n_in,
                              void* d_out, int out_size, void* d_ws, size_t ws_size,
                              hipStream_t stream)
{
  const float* x    = (const float*)d_in[0];
  const float* mask = (const float*)d_in[1];
  const float* W    = (const float*)d_in[2];
  const float* bias = (const float*)d_in[3];
  const float* cent = (const float*)d_in[4];
  float* out = (float*)d_out;

  float* ws   = (float*)d_ws;
  float* sa   = ws;                              // B*N*K = 524288
  float* invn = sa   + (size_t)B_ * N_ * K_;     // B*N   = 16384
  float* asum = invn + (size_t)B_ * N_;          // B*K   = 512
  float* vlad = asum + B_ * K_;                  // B*D*K = 262144
  float* eacc = vlad + (size_t)B_ * D_ * K_;     // 1
  float* msum = eacc + 1;                        // 1
  (void)in_sizes; (void)n_in; (void)out_size; (void)ws_size; (void)msum;

  // zero the accumulators each call (deterministic, graph-capture safe)
  (void)hipMemsetAsync(asum, 0, (B_ * K_) * sizeof(float), stream);
  (void)hipMemsetAsync(eacc, 0, 2 * sizeof(float), stream);

  netvlad_phase1<<<B_ * (N_ / 16), 512, 0, stream>>>(x, mask, W, bias,
                                                     sa, invn, asum, eacc, msum);
  netvlad_phase2<<<B_ * (D_ / 16), 64, 0, stream>>>(x, invn, sa, cent, asum, vlad);
  netvlad_phase3_norm<<<(B_ * K_) / 4, 128, 0, stream>>>(vlad, out);
  netvlad_phase3_dist<<<1, 512, 0, stream>>>(asum, eacc, msum, out);
}